// GDN_88304527606158
// MI455X (gfx1250) — compile-verified
//
#include <hip/hip_runtime.h>
#include <hip/hip_bf16.h>

// ---------------- problem constants ----------------
#define BATCH   8
#define CDIM    128
#define HWN     147456            // 384*384
#define KSLICES 32                // K-split for the Gram kernel
#define PXB     32                // pixels per workgroup block in norm kernel

#define REPARAM_OFFSET 0.0009765625f   // 2^-10

typedef __attribute__((ext_vector_type(2))) float v2f;
typedef __attribute__((ext_vector_type(4))) float v4f;
typedef __attribute__((ext_vector_type(8))) float v8f;

// raw GCC-vector int4 + address-space-qualified pointers for the async DMA builtin
typedef int v4i_raw __attribute__((vector_size(16)));
typedef __attribute__((address_space(1))) v4i_raw* as1_v4i_ptr;   // global
typedef __attribute__((address_space(3))) v4i_raw* as3_v4i_ptr;   // LDS

#if __has_builtin(__builtin_amdgcn_global_load_async_to_lds_b128)
#define ASYNC_STAGE 1
#else
#define ASYNC_STAGE 0
#endif

static __device__ __forceinline__ v8f wmma_f32(v2f a, v2f b, v8f c) {
    // D = A(16x4) * B(4x16) + C(16x16), fp32
    return __builtin_amdgcn_wmma_f32_16x16x4_f32(
        /*neg_a=*/false, a, /*neg_b=*/false, b,
        /*c_mod=*/(short)0, c, /*reuse_a=*/false, /*reuse_b=*/false);
}

// =====================================================================
// Kernel 1: per-batch Gram matrix  inner[b] = X[b] * X[b]^T   (C x C)
//   grid = BATCH * 4 * 4 * KSLICES blocks of 1 wave (32 threads)
//   each wave: 32x32 output block = 2x2 WMMA accumulators
//   K split KSLICES ways; partials merged via global atomic f32 add.
// =====================================================================
__global__ void __launch_bounds__(32)
gram_kernel(const float* __restrict__ X, float* __restrict__ inner)
{
    int lane = threadIdx.x & 31;
    int blk  = blockIdx.x;
    int ks = blk % KSLICES; blk /= KSLICES;
    int tj = blk & 3;       blk >>= 2;
    int ti = blk & 3;       blk >>= 2;
    int b  = blk;

    const int l15 = lane & 15;
    const int kh  = lane >> 4;            // 0 or 1 (K half: k+0/1 vs k+2/3)

    const size_t base = (size_t)b * CDIM * HWN;
    const float* pA0 = X + base + (size_t)(32 * ti + l15)      * HWN;
    const float* pA1 = X + base + (size_t)(32 * ti + l15 + 16) * HWN;
    const float* pB0 = X + base + (size_t)(32 * tj + l15)      * HWN;
    const float* pB1 = X + base + (size_t)(32 * tj + l15 + 16) * HWN;

    v8f d00 = {}, d01 = {}, d10 = {}, d11 = {};

    const int kBegin = ks * (HWN / KSLICES);
    const int kEnd   = kBegin + (HWN / KSLICES);

#pragma unroll 4
    for (int k = kBegin; k < kEnd; k += 4) {
        const int ko = k + 2 * kh;        // per-lane K offset; even -> 8B aligned
        v2f a0 = *(const v2f*)(pA0 + ko);
        v2f a1 = *(const v2f*)(pA1 + ko);
        v2f b0 = *(const v2f*)(pB0 + ko);
        v2f b1 = *(const v2f*)(pB1 + ko);
        d00 = wmma_f32(a0, b0, d00);
        d01 = wmma_f32(a0, b1, d01);
        d10 = wmma_f32(a1, b0, d10);
        d11 = wmma_f32(a1, b1, d11);
    }

    // D layout: lane l, VGPR v -> M = v + 8*(l>>4), N = l&15
    float* dst = inner + ((size_t)b * CDIM + 32 * ti) * CDIM + 32 * tj;
#pragma unroll
    for (int v = 0; v < 8; ++v) {
        const int m = v + 8 * kh;
        atomicAdd(dst + (size_t)m        * CDIM + l15,        d00[v]);
        atomicAdd(dst + (size_t)m        * CDIM + l15 + 16,   d01[v]);
        atomicAdd(dst + (size_t)(m + 16) * CDIM + l15,        d10[v]);
        atomicAdd(dst + (size_t)(m + 16) * CDIM + l15 + 16,   d11[v]);
    }
}

// =====================================================================
// Kernel 2: weight[b,i,j] = gamma_eff[i,j] * (1 + cos[b,i,j]),
//           beta_eff[i]   = max(beta[i], BOUND)^2 - offset
//   grid = BATCH blocks of 256 threads
// =====================================================================
__global__ void __launch_bounds__(256)
weight_kernel(const float* __restrict__ inner,
              const float* __restrict__ gamma,
              const float* __restrict__ beta,
              float* __restrict__ weight,
              float* __restrict__ beta_eff)
{
    const int b = blockIdx.x;
    const float BOUND = sqrtf(1e-6f + REPARAM_OFFSET);  // beta & gamma bounds equal
    const float* I = inner + (size_t)b * CDIM * CDIM;
    float* W = weight + (size_t)b * CDIM * CDIM;

    for (int idx = threadIdx.x; idx < CDIM * CDIM; idx += blockDim.x) {
        const int i = idx >> 7;
        const int j = idx & (CDIM - 1);
        const float inn = I[idx];
        const float ni = sqrtf(I[i * CDIM + i]);   // ||x_i||
        const float nj = sqrtf(I[j * CDIM + j]);   // ||x_j||
        const float cosv = inn / (ni * nj + 1e-6f);
        const float gp   = fmaxf(0.5f * (gamma[i * CDIM + j] + gamma[j * CDIM + i]), BOUND);
        const float geff = gp * gp - REPARAM_OFFSET;
        W[idx] = geff * (1.0f + cosv);
    }
    if (b == 0) {
        for (int i = threadIdx.x; i < CDIM; i += blockDim.x) {
            const float bp = fmaxf(beta[i], BOUND);
            beta_eff[i] = bp * bp - REPARAM_OFFSET;
        }
    }
}

// =====================================================================
// Kernel 3: norm_map = weight[b] @ |X[b]| ; out = x / (norm_map + beta_eff)
//   grid = BATCH * (HWN/PXB) blocks of 256 threads (8 waves)
//   Stage X[128 x 32px] block into LDS once (async DMA global->LDS when
//   available); wave w computes channels [16w, 16w+16) x 32 pixels with
//   64 WMMAs and a fused +beta / divide epilogue.
// =====================================================================
__global__ void __launch_bounds__(256)
norm_kernel(const float* __restrict__ X,
            const float* __restrict__ weight,
            const float* __restrict__ beta_eff,
            float* __restrict__ out)
{
    __shared__ float xs[CDIM][PXB];     // 16 KB, [channel][pixel]

    const int tid  = threadIdx.x;
    const int pb   = blockIdx.x % (HWN / PXB);
    const int b    = blockIdx.x / (HWN / PXB);
    const int px0  = pb * PXB;
    const size_t base = (size_t)b * CDIM * HWN;

    // ---- cooperative stage: 128 channels x 32 pixels = 1024 x 16B ----
#pragma unroll
    for (int r = 0; r < 4; ++r) {
        const int q   = tid + 256 * r;      // 0..1023
        const int c   = q >> 3;             // channel
        const int seg = q & 7;              // 4-pixel segment
        const float* gsrc = X + base + (size_t)c * HWN + px0 + seg * 4;
        float* ldst = &xs[c][seg * 4];
#if ASYNC_STAGE
        // direct global -> LDS DMA, tracked by ASYNCcnt (no VGPR round trip)
        __builtin_amdgcn_global_load_async_to_lds_b128(
            (as1_v4i_ptr)gsrc, (as3_v4i_ptr)ldst, /*offset=*/0, /*cpol=*/0);
#else
        *(v4f*)ldst = *(const v4f*)gsrc;
#endif
    }
#if ASYNC_STAGE
#if __has_builtin(__builtin_amdgcn_s_wait_asynccnt)
    __builtin_amdgcn_s_wait_asynccnt(0);
#else
    asm volatile("s_wait_asynccnt 0" ::: "memory");
#endif
#endif
    __syncthreads();

    const int lane  = tid & 31;
    const int wave  = tid >> 5;             // 0..7 -> channel tile
    const int mbase = 16 * wave;
    const int l15   = lane & 15;
    const int kh    = lane >> 4;

    // A = weight rows [mbase .. mbase+15], K = 0..127 (L2-resident)
    const float* Arow = weight + ((size_t)b * CDIM + mbase + l15) * CDIM;

    v8f d0 = {}, d1 = {};
#pragma unroll
    for (int k = 0; k < CDIM; k += 4) {
        const int kr = k + 2 * kh;
        v2f a  = *(const v2f*)(Arow + kr);
        v2f b0 = { fabsf(xs[kr][l15]),      fabsf(xs[kr + 1][l15])      };
        v2f b1 = { fabsf(xs[kr][l15 + 16]), fabsf(xs[kr + 1][l15 + 16]) };
        d0 = wmma_f32(a, b0, d0);
        d1 = wmma_f32(a, b1, d1);
    }

    // ---- fused epilogue: out = x / (norm + beta_eff) ----
#pragma unroll
    for (int v = 0; v < 8; ++v) {
        const int m  = mbase + v + 8 * kh;        // output channel
        const float be = beta_eff[m];
        const float x0 = xs[m][l15];
        const float x1 = xs[m][l15 + 16];
        float* orow = out + ((size_t)b * CDIM + m) * HWN + px0;
        orow[l15]      = x0 / (d0[v] + be);
        orow[l15 + 16] = x1 / (d1[v] + be);
    }
}

// =====================================================================
extern "C" void kernel_launch(void* const* d_in, const int* in_sizes, int n_in,
                              void* d_out, int out_size, void* d_ws, size_t ws_size,
                              hipStream_t stream) {
    const float* X     = (const float*)d_in[0];   // [8,128,384,384]
    const float* beta  = (const float*)d_in[1];   // [128]
    const float* gamma = (const float*)d_in[2];   // [128,128]
    float* out = (float*)d_out;

    float* inner    = (float*)d_ws;                       // [8,128,128]
    float* weight   = inner  + BATCH * CDIM * CDIM;       // [8,128,128]
    float* beta_eff = weight + BATCH * CDIM * CDIM;       // [128]

    (void)hipMemsetAsync(inner, 0, (size_t)BATCH * CDIM * CDIM * sizeof(float), stream);

    gram_kernel<<<BATCH * 4 * 4 * KSLICES, 32, 0, stream>>>(X, inner);
    weight_kernel<<<BATCH, 256, 0, stream>>>(inner, gamma, beta, weight, beta_eff);
    norm_kernel<<<BATCH * (HWN / PXB), 256, 0, stream>>>(X, weight, beta_eff, out);
}